// GAT_31834297598575
// MI455X (gfx1250) — compile-verified
//
#include <hip/hip_runtime.h>
#include <math.h>

// ---------------------------------------------------------------------------
// GATv2 (3 layers) + scatter-mean pooling for MI455X (gfx1250, wave32).
// Dense h@W transforms use V_WMMA_F32_16X16X4_F32 with a branch-free inner
// loop (K and N padded, LDB/LDC compile-time -> immediate-offset loads),
// 16x32 tile per wave -> 4 WMMAs per A fetch. Edge softmax/aggregation are
// the memory-bound path: b128 vector loads + f32 global atomics (dest
// buffers L2-resident: 41 MB << 192 MB L2).
// ---------------------------------------------------------------------------

typedef __attribute__((ext_vector_type(2))) float v2f;
typedef __attribute__((ext_vector_type(8))) float v8f;

#define NNODES 40000
#define NEDGES 500000
#define NGRAPH 512
#define NEG_SLOPE 0.2f
#define FMAXF 3.402823466e+38f

static __device__ __forceinline__ void atomicMaxF(float* addr, float val) {
  // float max via monotonic int/uint trick (init buffer to -FLT_MAX).
  if (__float_as_int(val) >= 0)
    atomicMax((int*)addr, __float_as_int(val));
  else
    atomicMin((unsigned int*)addr, __float_as_uint(val));
}

// ---------------- utility fills / pads -------------------------------------
__global__ void fill_f32(float* __restrict__ p, float v, int n) {
  int i = blockIdx.x * blockDim.x + threadIdx.x;
  if (i < n) p[i] = v;
}

// dst[Kp x Ncp] = zero-padded src[K x Nc]
__global__ void pad_weights_kernel(const float* __restrict__ src, float* __restrict__ dst,
                                   int K, int Nc, int Kp, int Ncp) {
  int i = blockIdx.x * blockDim.x + threadIdx.x;
  if (i >= Kp * Ncp) return;
  int k = i / Ncp, c = i - k * Ncp;
  dst[i] = (k < K && c < Nc) ? src[k * Nc + c] : 0.0f;
}

// ---------------- self-loop attr (mean of incoming edge_attr) --------------
__global__ void edge_deg_kernel(const int* __restrict__ dst,
                                const float* __restrict__ ea,
                                float* __restrict__ deg,
                                float* __restrict__ esum, int E) {
  int e = blockIdx.x * blockDim.x + threadIdx.x;
  if (e >= E) return;
  int d = dst[e];
  atomicAdd(&deg[d], 1.0f);
  atomicAdd(&esum[d], ea[e]);
}

__global__ void loop_attr_kernel(const float* __restrict__ deg,
                                 const float* __restrict__ esum,
                                 float* __restrict__ loopa, int n) {
  int i = blockIdx.x * blockDim.x + threadIdx.x;
  if (i >= n) return;
  float dg = deg[i];
  loopa[i] = (dg > 0.0f) ? esum[i] / fmaxf(dg, 1.0f) : 0.0f;
}

// ------- embedding: [x0@w0+b0, x1:4, 0pad] -> 8 features (K padded) --------
__global__ void embed_kernel(const float* __restrict__ x,
                             const float* __restrict__ w0,
                             const float* __restrict__ b0,
                             float* __restrict__ h, int n) {
  int i = blockIdx.x * blockDim.x + threadIdx.x;
  if (i >= n) return;
  float t = x[i * 4 + 0];
  #pragma unroll
  for (int j = 0; j < 4; ++j) h[i * 8 + j] = t * w0[j] + b0[j];
  #pragma unroll
  for (int j = 0; j < 3; ++j) h[i * 8 + 4 + j] = x[i * 4 + 1 + j];
  h[i * 8 + 7] = 0.0f;
}

// ---------------- fused dual-B WMMA GEMM (C1 = A@B1, C2 = A@B2) ------------
// fp32 WMMA 16x16x4. One wave computes a 16x32 output tile for BOTH B's:
// 4 WMMAs per A fetch, branch-free inner loop. LDB is compile-time and
// equals both the padded column count (Ncp) and the C row stride, so every
// B load / C store uses a constant immediate offset.
// Requires: M%16==0, K%4==0, LDB%32==0 (caller pads).
template <int LDB>
__global__ __launch_bounds__(256)
void gemm2_wmma_kernel(const float* __restrict__ A, int lda,
                       const float* __restrict__ B1,
                       const float* __restrict__ B2,
                       float* __restrict__ C1, float* __restrict__ C2,
                       int M, int K) {
  int lane = threadIdx.x & 31;
  int wave = (blockIdx.x * blockDim.x + threadIdx.x) >> 5;
  constexpr int tn_count = LDB >> 5;
  int tm_count = M >> 4;
  if (wave >= tm_count * tn_count) return;   // wave-uniform: EXEC stays all-1s
  int tm = wave / tn_count;
  int tn = wave - tm * tn_count;

  int l    = lane & 15;
  int khi  = (lane >> 4) << 1;     // lanes 0-15 -> K0/K1, lanes 16-31 -> K2/K3
  int row  = tm * 16 + l;
  int col0 = tn * 32 + l;
  const float* Arow = A + (size_t)row * lda + khi;   // (lda,khi even -> 8B aligned)
  const float* b1p  = B1 + (size_t)khi * LDB + col0;
  const float* b2p  = B2 + (size_t)khi * LDB + col0;

  v8f c1a = {}; v8f c1b = {}; v8f c2a = {}; v8f c2b = {};
  #pragma unroll 2
  for (int kk = 0; kk < K; kk += 4) {
    v2f a = *(const v2f*)(Arow);                      // global_load_b64
    Arow += 4;
    v2f b1a = { b1p[0],  b1p[LDB] };                  // imm-offset b32 loads
    v2f b1b = { b1p[16], b1p[LDB + 16] };
    v2f b2a = { b2p[0],  b2p[LDB] };
    v2f b2b = { b2p[16], b2p[LDB + 16] };
    b1p += 4 * LDB;
    b2p += 4 * LDB;
    // (neg_a, A, neg_b, B, c_mod, C, reuse_a, reuse_b)
    c1a = __builtin_amdgcn_wmma_f32_16x16x4_f32(false, a, false, b1a, (short)0, c1a, false, false);
    c1b = __builtin_amdgcn_wmma_f32_16x16x4_f32(false, a, false, b1b, (short)0, c1b, false, false);
    c2a = __builtin_amdgcn_wmma_f32_16x16x4_f32(false, a, false, b2a, (short)0, c2a, false, false);
    c2b = __builtin_amdgcn_wmma_f32_16x16x4_f32(false, a, false, b2b, (short)0, c2b, false, false);
  }

  int orow0 = tm * 16 + ((lane >> 4) << 3);  // C layout: VGPR r -> row r (+8 hi lanes)
  int ocol  = tn * 32 + l;
  float* c1o = C1 + (size_t)orow0 * LDB + ocol;
  float* c2o = C2 + (size_t)orow0 * LDB + ocol;
  #pragma unroll
  for (int r = 0; r < 8; ++r) {              // imm-offset stores (r*LDB*4 <= 7168B)
    c1o[r * LDB]      = c1a[r];
    c1o[r * LDB + 16] = c1b[r];
    c2o[r * LDB]      = c2a[r];
    c2o[r * LDB + 16] = c2b[r];
  }
}

// ---------------- edge helpers ---------------------------------------------
static __device__ __forceinline__ void edge_fetch(int e, const int* SRC, const int* DST,
                                                  const float* EA, const float* LOOP,
                                                  int E, int& s, int& d, float& a) {
  if (e < E) { s = SRC[e]; d = DST[e]; a = EA[e]; }
  else       { s = d = e - E; a = LOOP[e - E]; }
}

static __device__ __forceinline__ void edge_fetch_sd(int e, const int* SRC, const int* DST,
                                                     int E, int& s, int& d) {
  if (e < E) { s = SRC[e]; d = DST[e]; }
  else       { s = d = e - E; }
}

static __device__ __forceinline__ float lrelu_dot(float4 vl, float4 vr, float4 vw,
                                                  float4 va, float a) {
  float p = 0.0f, v;
  v = vl.x + vr.x + a * vw.x; v = (v > 0.0f) ? v : NEG_SLOPE * v; p += v * va.x;
  v = vl.y + vr.y + a * vw.y; v = (v > 0.0f) ? v : NEG_SLOPE * v; p += v * va.y;
  v = vl.z + vr.z + a * vw.z; v = (v > 0.0f) ? v : NEG_SLOPE * v; p += v * va.z;
  v = vl.w + vr.w + a * vw.w; v = (v > 0.0f) ? v : NEG_SLOPE * v; p += v * va.w;
  return p;
}

// ---------------- edge pass 1: logits + segment max (wave per edge) --------
template <int H, int C>
__global__ __launch_bounds__(256)
void edge_logits_kernel(const float* __restrict__ xl, const float* __restrict__ xr,
                        const float* __restrict__ we, const float* __restrict__ att,
                        const int* __restrict__ SRC, const int* __restrict__ DST,
                        const float* __restrict__ EA, const float* __restrict__ LOOP,
                        float* __restrict__ logits, float* __restrict__ mbuf,
                        int E, int EF) {
  constexpr int F = H * C;
  constexpr int PER = F / 32;     // channels per lane (8 for F=256, 4 for F=128)
  constexpr int NV = PER / 4;     // float4 loads per lane
  constexpr int G = 32 / H;       // lanes per head (4)
  int lane = threadIdx.x & 31;
  int e = (blockIdx.x * blockDim.x + threadIdx.x) >> 5;
  if (e >= EF) return;
  int s, d; float a;
  edge_fetch(e, SRC, DST, EA, LOOP, E, s, d, a);
  int f0 = lane * PER;
  const float4* xls = (const float4*)(xl + (size_t)s * F + f0);
  const float4* xrs = (const float4*)(xr + (size_t)d * F + f0);
  const float4* wes = (const float4*)(we + f0);
  const float4* ats = (const float4*)(att + f0);
  float partial = 0.0f;
  #pragma unroll
  for (int j = 0; j < NV; ++j)
    partial += lrelu_dot(xls[j], xrs[j], wes[j], ats[j], a);
  #pragma unroll
  for (int off = 1; off < G; off <<= 1)
    partial += __shfl_xor(partial, off, 32);
  if ((lane & (G - 1)) == 0) {
    int h = lane / G;
    logits[(size_t)e * H + h] = partial;
    atomicMaxF(&mbuf[(size_t)d * H + h], partial);
  }
}

// scalar variant for layer 3 (H=1, C=8); FS = stride of xl/xr rows
__global__ void edge_logits8_kernel(const float* __restrict__ xl, const float* __restrict__ xr,
                                    const float* __restrict__ we, const float* __restrict__ att,
                                    const int* __restrict__ SRC, const int* __restrict__ DST,
                                    const float* __restrict__ EA, const float* __restrict__ LOOP,
                                    float* __restrict__ logits, float* __restrict__ mbuf,
                                    int E, int EF, int FS) {
  int e = blockIdx.x * blockDim.x + threadIdx.x;
  if (e >= EF) return;
  int s, d; float a;
  edge_fetch(e, SRC, DST, EA, LOOP, E, s, d, a);
  const float4* xls = (const float4*)(xl + (size_t)s * FS);
  const float4* xrs = (const float4*)(xr + (size_t)d * FS);
  const float4* wes = (const float4*)we;
  const float4* ats = (const float4*)att;
  float lg = lrelu_dot(xls[0], xrs[0], wes[0], ats[0], a)
           + lrelu_dot(xls[1], xrs[1], wes[1], ats[1], a);
  logits[e] = lg;
  atomicMaxF(&mbuf[d], lg);
}

// ---------------- edge pass 2: exp + segment sum (thread per edge*head) ----
template <int H>
__global__ void edge_expsum_kernel(float* __restrict__ logits, const float* __restrict__ mbuf,
                                   float* __restrict__ denom, const int* __restrict__ DST,
                                   int E, int EF) {
  int idx = blockIdx.x * blockDim.x + threadIdx.x;
  if (idx >= EF * H) return;
  int e = idx / H, h = idx - e * H;
  int d = (e < E) ? DST[e] : e - E;
  float ex = __expf(logits[idx] - mbuf[d * H + h]);
  logits[idx] = ex;               // overwrite logit with exp value
  atomicAdd(&denom[d * H + h], ex);
}

// ---------------- edge pass 3: alpha * xl[src] scatter (wave per edge) -----
template <int H, int C>
__global__ __launch_bounds__(256)
void edge_aggr_kernel(const float* __restrict__ xl, const float* __restrict__ logits,
                      const float* __restrict__ denom, float* __restrict__ out,
                      const int* __restrict__ SRC, const int* __restrict__ DST,
                      int E, int EF) {
  constexpr int F = H * C;
  constexpr int PER = F / 32;
  constexpr int NV = PER / 4;
  constexpr int G = 32 / H;
  int lane = threadIdx.x & 31;
  int e = (blockIdx.x * blockDim.x + threadIdx.x) >> 5;
  if (e >= EF) return;
  int s, d;
  edge_fetch_sd(e, SRC, DST, E, s, d);
  int h = lane / G;   // all PER channels of this lane belong to one head
  float alpha = logits[(size_t)e * H + h] / denom[(size_t)d * H + h];
  int f0 = lane * PER;
  const float4* xls = (const float4*)(xl + (size_t)s * F + f0);
  float* outp = out + (size_t)d * F + f0;
  #pragma unroll
  for (int j = 0; j < NV; ++j) {
    float4 v = xls[j];
    atomicAdd(&outp[4 * j + 0], alpha * v.x);
    atomicAdd(&outp[4 * j + 1], alpha * v.y);
    atomicAdd(&outp[4 * j + 2], alpha * v.z);
    atomicAdd(&outp[4 * j + 3], alpha * v.w);
  }
}

__global__ void edge_aggr8_kernel(const float* __restrict__ xl, const float* __restrict__ logits,
                                  const float* __restrict__ denom, float* __restrict__ out,
                                  const int* __restrict__ SRC, const int* __restrict__ DST,
                                  int E, int EF, int FS) {
  int e = blockIdx.x * blockDim.x + threadIdx.x;
  if (e >= EF) return;
  int s, d;
  edge_fetch_sd(e, SRC, DST, E, s, d);
  float alpha = logits[e] / denom[d];
  const float4* xls = (const float4*)(xl + (size_t)s * FS);
  float4 v0 = xls[0], v1 = xls[1];
  float* outp = out + (size_t)d * 8;
  atomicAdd(&outp[0], alpha * v0.x);
  atomicAdd(&outp[1], alpha * v0.y);
  atomicAdd(&outp[2], alpha * v0.z);
  atomicAdd(&outp[3], alpha * v0.w);
  atomicAdd(&outp[4], alpha * v1.x);
  atomicAdd(&outp[5], alpha * v1.y);
  atomicAdd(&outp[6], alpha * v1.z);
  atomicAdd(&outp[7], alpha * v1.w);
}

// ---------------- bias + tanh (in stride ldin -> packed out stride F) ------
__global__ void bias_tanh_kernel(const float* __restrict__ in, int ldin,
                                 const float* __restrict__ bias,
                                 float* __restrict__ out, int F, int nTotal) {
  int i = blockIdx.x * blockDim.x + threadIdx.x;
  if (i >= nTotal) return;
  int node = i / F, f = i - node * F;
  out[i] = tanhf(in[(size_t)node * ldin + f] + bias[f]);
}

// ---------------- scatter-mean pooling + final linear ----------------------
__global__ void pool_kernel(const float* __restrict__ h, const int* __restrict__ batch,
                            float* __restrict__ pooled, float* __restrict__ cnt, int n) {
  int i = blockIdx.x * blockDim.x + threadIdx.x;
  if (i >= n) return;
  int b = batch[i];
  const float4* hp = (const float4*)(h + (size_t)i * 8);
  float4 v0 = hp[0], v1 = hp[1];
  float* pp = pooled + b * 8;
  atomicAdd(&pp[0], v0.x); atomicAdd(&pp[1], v0.y);
  atomicAdd(&pp[2], v0.z); atomicAdd(&pp[3], v0.w);
  atomicAdd(&pp[4], v1.x); atomicAdd(&pp[5], v1.y);
  atomicAdd(&pp[6], v1.z); atomicAdd(&pp[7], v1.w);
  atomicAdd(&cnt[b], 1.0f);
}

__global__ void final_kernel(const float* __restrict__ pooled, const float* __restrict__ cnt,
                             const float* __restrict__ w4, const float* __restrict__ b4,
                             float* __restrict__ out, int nb) {
  int b = blockIdx.x * blockDim.x + threadIdx.x;
  if (b >= nb) return;
  float cc = fmaxf(cnt[b], 1.0f);
  float s = 0.0f;
  #pragma unroll
  for (int c = 0; c < 8; ++c) s += (pooled[b * 8 + c] / cc) * w4[c];
  out[b] = s + b4[0];
}

// ---------------------------------------------------------------------------
extern "C" void kernel_launch(void* const* d_in, const int* in_sizes, int n_in,
                              void* d_out, int out_size, void* d_ws, size_t ws_size,
                              hipStream_t stream) {
  const float* x     = (const float*)d_in[0];
  const int*   ei    = (const int*)  d_in[1];
  const float* ea    = (const float*)d_in[2];
  const int*   batch = (const int*)  d_in[3];
  const float* w0  = (const float*)d_in[4];
  const float* b0  = (const float*)d_in[5];
  const float* wl1 = (const float*)d_in[6];
  const float* wr1 = (const float*)d_in[7];
  const float* we1 = (const float*)d_in[8];
  const float* at1 = (const float*)d_in[9];
  const float* bb1 = (const float*)d_in[10];
  const float* wl2 = (const float*)d_in[11];
  const float* wr2 = (const float*)d_in[12];
  const float* we2 = (const float*)d_in[13];
  const float* at2 = (const float*)d_in[14];
  const float* bb2 = (const float*)d_in[15];
  const float* wl3 = (const float*)d_in[16];
  const float* wr3 = (const float*)d_in[17];
  const float* we3 = (const float*)d_in[18];
  const float* at3 = (const float*)d_in[19];
  const float* bb3 = (const float*)d_in[20];
  const float* w4  = (const float*)d_in[21];
  const float* b4  = (const float*)d_in[22];

  const int N  = NNODES;
  const int E  = NEDGES;
  const int EF = NEDGES + NNODES;  // edges + self loops
  const int*   SRC = ei;
  const int*   DST = ei + E;

  // workspace layout (floats); ~144 MB total
  float* ws     = (float*)d_ws;
  float* hbuf   = ws;                              // N*256  node features
  float* xl     = hbuf   + (size_t)N * 256;        // N*256
  float* xout   = xl     + (size_t)N * 256;        // N*256  xr, then recycled as out
  float* logits = xout   + (size_t)N * 256;        // EF*8
  float* mbuf   = logits + (size_t)EF * 8;         // N*8
  float* denom  = mbuf   + (size_t)N * 8;          // N*8
  float* deg    = denom  + (size_t)N * 8;          // N
  float* esum   = deg    + N;                      // N
  float* loopa  = esum   + N;                      // N
  float* pooled = loopa  + N;                      // B*8
  float* cnt    = pooled + NGRAPH * 8;             // B
  float* wp1l   = cnt    + NGRAPH;                 // 8*256   padded wl1
  float* wp1r   = wp1l   + 8 * 256;                // 8*256   padded wr1
  float* wp3l   = wp1r   + 8 * 256;                // 128*32  padded wl3
  float* wp3r   = wp3l   + 128 * 32;               // 128*32  padded wr3

  const int T = 256;
  auto cdiv = [](long long a, long long b) { return (int)((a + b - 1) / b); };

  // ---- weight padding (K pad for layer1, N pad for layer3) ----
  pad_weights_kernel<<<cdiv(8 * 256, T), T, 0, stream>>>(wl1, wp1l, 7, 256, 8, 256);
  pad_weights_kernel<<<cdiv(8 * 256, T), T, 0, stream>>>(wr1, wp1r, 7, 256, 8, 256);
  pad_weights_kernel<<<cdiv(128 * 32, T), T, 0, stream>>>(wl3, wp3l, 128, 8, 128, 32);
  pad_weights_kernel<<<cdiv(128 * 32, T), T, 0, stream>>>(wr3, wp3r, 128, 8, 128, 32);

  // ---- self-loop attributes ----
  fill_f32<<<cdiv(N, T), T, 0, stream>>>(deg, 0.0f, N);
  fill_f32<<<cdiv(N, T), T, 0, stream>>>(esum, 0.0f, N);
  edge_deg_kernel<<<cdiv(E, T), T, 0, stream>>>(DST, ea, deg, esum, E);
  loop_attr_kernel<<<cdiv(N, T), T, 0, stream>>>(deg, esum, loopa, N);

  // ---- embedding -> h (N x 8, col 7 zero-padded) ----
  embed_kernel<<<cdiv(N, T), T, 0, stream>>>(x, w0, b0, hbuf, N);

  // ================= Layer 1: 7 -> 256 (H=8, C=32) =================
  {
    const int K = 8, F = 256, H = 8;
    int waves = (N >> 4) * (F >> 5);
    gemm2_wmma_kernel<256><<<cdiv((long long)waves * 32, T), T, 0, stream>>>(
        hbuf, 8, wp1l, wp1r, xl, xout, N, K);
    fill_f32<<<cdiv(N * H, T), T, 0, stream>>>(mbuf, -FMAXF, N * H);
    fill_f32<<<cdiv(N * H, T), T, 0, stream>>>(denom, 0.0f, N * H);
    edge_logits_kernel<8, 32><<<cdiv((long long)EF * 32, T), T, 0, stream>>>(
        xl, xout, we1, at1, SRC, DST, ea, loopa, logits, mbuf, E, EF);
    edge_expsum_kernel<8><<<cdiv((long long)EF * H, T), T, 0, stream>>>(
        logits, mbuf, denom, DST, E, EF);
    fill_f32<<<cdiv((long long)N * F, T), T, 0, stream>>>(xout, 0.0f, N * F);
    edge_aggr_kernel<8, 32><<<cdiv((long long)EF * 32, T), T, 0, stream>>>(
        xl, logits, denom, xout, SRC, DST, E, EF);
    bias_tanh_kernel<<<cdiv((long long)N * F, T), T, 0, stream>>>(
        xout, F, bb1, hbuf, F, N * F);
  }

  // ================= Layer 2: 256 -> 128 (H=8, C=16) =================
  {
    const int K = 256, F = 128, H = 8;
    int waves = (N >> 4) * (F >> 5);
    gemm2_wmma_kernel<128><<<cdiv((long long)waves * 32, T), T, 0, stream>>>(
        hbuf, K, wl2, wr2, xl, xout, N, K);
    fill_f32<<<cdiv(N * H, T), T, 0, stream>>>(mbuf, -FMAXF, N * H);
    fill_f32<<<cdiv(N * H, T), T, 0, stream>>>(denom, 0.0f, N * H);
    edge_logits_kernel<8, 16><<<cdiv((long long)EF * 32, T), T, 0, stream>>>(
        xl, xout, we2, at2, SRC, DST, ea, loopa, logits, mbuf, E, EF);
    edge_expsum_kernel<8><<<cdiv((long long)EF * H, T), T, 0, stream>>>(
        logits, mbuf, denom, DST, E, EF);
    fill_f32<<<cdiv((long long)N * F, T), T, 0, stream>>>(xout, 0.0f, N * F);
    edge_aggr_kernel<8, 16><<<cdiv((long long)EF * 32, T), T, 0, stream>>>(
        xl, logits, denom, xout, SRC, DST, E, EF);
    bias_tanh_kernel<<<cdiv((long long)N * F, T), T, 0, stream>>>(
        xout, F, bb2, hbuf, F, N * F);
  }

  // ================= Layer 3: 128 -> 8 (H=1, C=8), Nc padded to 32 ========
  {
    const int K = 128, Fp = 32;   // padded GEMM output stride
    int waves = (N >> 4) * (Fp >> 5);
    gemm2_wmma_kernel<32><<<cdiv((long long)waves * 32, T), T, 0, stream>>>(
        hbuf, K, wp3l, wp3r, xl, xout, N, K);
    fill_f32<<<cdiv(N, T), T, 0, stream>>>(mbuf, -FMAXF, N);
    fill_f32<<<cdiv(N, T), T, 0, stream>>>(denom, 0.0f, N);
    edge_logits8_kernel<<<cdiv(EF, T), T, 0, stream>>>(
        xl, xout, we3, at3, SRC, DST, ea, loopa, logits, mbuf, E, EF, Fp);
    edge_expsum_kernel<1><<<cdiv(EF, T), T, 0, stream>>>(
        logits, mbuf, denom, DST, E, EF);
    // hbuf is free after the GEMM -> use it as the aggregation accumulator
    // (xl, stride 32, is still being read during aggregation).
    fill_f32<<<cdiv(N * 8, T), T, 0, stream>>>(hbuf, 0.0f, N * 8);
    edge_aggr8_kernel<<<cdiv(EF, T), T, 0, stream>>>(
        xl, logits, denom, hbuf, SRC, DST, E, EF, Fp);
    // bias+tanh: hbuf(packed 8) -> xout(packed 8)
    bias_tanh_kernel<<<cdiv(N * 8, T), T, 0, stream>>>(
        hbuf, 8, bb3, xout, 8, N * 8);
  }

  // ---- scatter-mean pooling + final linear ----
  fill_f32<<<cdiv(NGRAPH * 8, T), T, 0, stream>>>(pooled, 0.0f, NGRAPH * 8);
  fill_f32<<<cdiv(NGRAPH, T), T, 0, stream>>>(cnt, 0.0f, NGRAPH);
  pool_kernel<<<cdiv(N, T), T, 0, stream>>>(xout, batch, pooled, cnt, N);
  final_kernel<<<cdiv(NGRAPH, T), T, 0, stream>>>(pooled, cnt, w4, b4,
                                                  (float*)d_out, NGRAPH);
}